// Model_88433376625452
// MI455X (gfx1250) — compile-verified
//
#include <hip/hip_runtime.h>

#define LOG2PI_F 1.8378770664093453f

#define NBLOCKS 512
#define TPB     1024

// LDS staging capacities (floats)
#define CAP_TG   20000
#define CAP_TFH  1000
#define SOFF_TGMU   0
#define SOFF_TGSIG  (CAP_TG)
#define SOFF_TFMU   (2 * CAP_TG)
#define SOFF_TFSIG  (2 * CAP_TG + CAP_TFH)
#define SOFF_RED    (2 * CAP_TG + 2 * CAP_TFH)
#define S_TOTAL     (SOFF_RED + 32)          // 42032 floats = 168128 B < 320 KB

typedef __attribute__((ext_vector_type(2))) float        v2f;
typedef __attribute__((ext_vector_type(8))) float        v8f;
typedef __attribute__((ext_vector_type(4))) unsigned int v4u;
typedef __attribute__((ext_vector_type(8))) int          v8i;
typedef __attribute__((ext_vector_type(4))) int          v4i;

__device__ __forceinline__ int imin(int a, int b) { return a < b ? a : b; }

// ---------------------------------------------------------------------------
// Exact-f32 wave32 sum via V_WMMA_F32_16X16X4_F32.
// A: VGPR0 = s (A[m][0]=s_m lanes 0-15, A[m][2]=s_{m+16} lanes 16-31),
//    VGPR1 = 0. B = all ones. D[m][n] = s_m + s_{m+16} = r_m.
// Per-lane sum of D's 8 VGPRs gives sum(r_0..r_7) (lanes<16) or
// sum(r_8..r_15) (lanes>=16); one ds_swizzle SWAPX16 + add completes the
// 32-lane total. Requires EXEC == all ones across the executing wave.
// ---------------------------------------------------------------------------
__device__ __forceinline__ float wave_reduce_wmma(float s) {
    v2f a; a[0] = s;    a[1] = 0.0f;
    v2f b; b[0] = 1.0f; b[1] = 1.0f;
    v8f c = {};
    v8f d = __builtin_amdgcn_wmma_f32_16x16x4_f32(
        /*neg_a=*/false, a, /*neg_b=*/false, b,
        /*c_mod=*/(short)0, c, /*reuse_a=*/false, /*reuse_b=*/false);
    float t = ((d[0] + d[1]) + (d[2] + d[3])) + ((d[4] + d[5]) + (d[6] + d[7]));
    // SWAPX16: ds_swizzle group-of-32, and_mask=0x1f, or_mask=0, xor_mask=0x10
    int sw = __builtin_amdgcn_ds_swizzle(__float_as_int(t), 0x401f);
    return t + __int_as_float(sw);
}

// ---------------------------------------------------------------------------
// Tensor Data Mover: 1D copy of n 4-byte elements global -> LDS.
// Descriptor per CDNA5 ISA ch.8 (group0: count=1, lds_addr, global_addr,
// type=2; group1: data_size=4B, tensor_dim0=tile_dim0=n, dim1=1, stride=n).
// 6-arg builtin on this toolchain; trailing groups unused for a 1-D tile.
// Issue from ONE wave only; EXEC is ignored by TENSOR_LOAD_TO_LDS.
// ---------------------------------------------------------------------------
__device__ __forceinline__ void tdm_load_1d(const void* gptr,
                                            unsigned lds_byte_off,
                                            unsigned n) {
    unsigned long long ga = (unsigned long long)gptr;
    v4u g0;
    g0[0] = 1u;                                            // count=1, no gather
    g0[1] = lds_byte_off;                                  // lds_addr
    g0[2] = (unsigned)(ga & 0xFFFFFFFFu);                  // global_addr[31:0]
    g0[3] = (unsigned)((ga >> 32) & 0x01FFFFFFu) | (2u << 30); // addr[56:32] | type=2
    v8i g1;
    g1[0] = (int)(2u << 16);                               // data_size=2 (4B), mask=0
    g1[1] = (int)((n & 0xFFFFu) << 16);                    // tensor_dim0[15:0] @ bit48
    g1[2] = (int)((n >> 16) | (1u << 16));                 // tensor_dim0[31:16], tensor_dim1=1
    g1[3] = (int)((n & 0xFFFFu) << 16);                    // tile_dim0 @ bit112
    g1[4] = 1;                                             // tile_dim1=1, tile_dim2=0
    g1[5] = (int)n;                                        // tensor_dim0_stride[31:0]
    g1[6] = 0;
    g1[7] = 0;
    v4i gz4 = {0, 0, 0, 0};
    v8i gz8 = {0, 0, 0, 0, 0, 0, 0, 0};
    __builtin_amdgcn_tensor_load_to_lds(g0, g1, gz4, gz4, gz8, 0);
}

// Per-edge conditional-Gaussian log-prob (sqrt eliminated: log(scale)=0.5*log(var)).
// TF-parameter gather is done by *index selection* into the single LDS arena
// so each edge costs exactly 4 ds_load_b32 (tf_mu, tf_sig, tg_mu, tg_sig).
__device__ __forceinline__ float edge_lp(float k, float al, float cv,
                                         float ey, float ex,
                                         int it, int ih, int itg, int hi,
                                         const float* __restrict__ lds) {
    int mu_idx  = hi ? (SOFF_TFMU + ih) : it;      // TG mu table is at offset 0
    int sig_add = hi ? CAP_TFH : CAP_TG;           // sigma table follows its mu table
    float tf_mu  = lds[mu_idx];
    float tf_sig = lds[mu_idx + sig_add];
    float tg_mu  = lds[itg];
    float tg_sig = lds[itg + CAP_TG];
    float inv_var = __builtin_amdgcn_rcpf(tf_sig * tf_sig);
    float loc = tg_mu + k * cv * (ey - tf_mu) * inv_var;
    loc = fmaxf(loc, 0.0f) + 0.01f;
    float var = tg_sig * tg_sig - al * al * inv_var;
    var = fmaxf(var, 0.0f) + 0.01f;
    float diff = ex - loc;
    return -0.5f * (diff * diff * __builtin_amdgcn_rcpf(var) + __logf(var) + LOG2PI_F);
}

__global__ __launch_bounds__(TPB) void edge_kernel(
    const float* __restrict__ TGmu,  const float* __restrict__ TGsig,
    const float* __restrict__ TFmu,  const float* __restrict__ TFsig,
    const float* __restrict__ k_edge, const float* __restrict__ alpha,
    const float* __restrict__ cov,    const float* __restrict__ edge_y,
    const float* __restrict__ edge_x,
    const int* __restrict__ idx_tf_tg, const int* __restrict__ idx_tf_high,
    const int* __restrict__ edge_tg,   const int* __restrict__ is_high,
    float* __restrict__ partials, int nE, int nTG, int nTFH) {
    __shared__ float s_all[S_TOTAL];
    const int tid = threadIdx.x;

    // Wave 0 issues the four TDM descriptors, waits, then the WG barrier
    // publishes the staged tables to all waves.
    if (tid < 32) {
        tdm_load_1d(TGmu,  SOFF_TGMU  * 4u, (unsigned)imin(nTG,  CAP_TG));
        tdm_load_1d(TGsig, SOFF_TGSIG * 4u, (unsigned)imin(nTG,  CAP_TG));
        tdm_load_1d(TFmu,  SOFF_TFMU  * 4u, (unsigned)imin(nTFH, CAP_TFH));
        tdm_load_1d(TFsig, SOFF_TFSIG * 4u, (unsigned)imin(nTFH, CAP_TFH));
        __builtin_amdgcn_s_wait_tensorcnt(0);
    }
    __syncthreads();

    const float* lds = s_all;

    float acc = 0.0f;
    const int gid    = blockIdx.x * TPB + tid;
    const int stride = gridDim.x * TPB;
    const int n4     = nE >> 2;

    for (int g = gid; g < n4; g += stride) {
        float4 k  = ((const float4*)k_edge)[g];
        float4 al = ((const float4*)alpha)[g];
        float4 cv = ((const float4*)cov)[g];
        float4 ey = ((const float4*)edge_y)[g];
        float4 ex = ((const float4*)edge_x)[g];
        int4   it = ((const int4*)idx_tf_tg)[g];
        int4   ih = ((const int4*)idx_tf_high)[g];
        int4   tg = ((const int4*)edge_tg)[g];
        int4   hb = ((const int4*)is_high)[g];
        acc += edge_lp(k.x, al.x, cv.x, ey.x, ex.x, it.x, ih.x, tg.x, hb.x, lds);
        acc += edge_lp(k.y, al.y, cv.y, ey.y, ex.y, it.y, ih.y, tg.y, hb.y, lds);
        acc += edge_lp(k.z, al.z, cv.z, ey.z, ex.z, it.z, ih.z, tg.z, hb.z, lds);
        acc += edge_lp(k.w, al.w, cv.w, ey.w, ex.w, it.w, ih.w, tg.w, hb.w, lds);
    }
    for (int i = (n4 << 2) + gid; i < nE; i += stride) {
        acc += edge_lp(k_edge[i], alpha[i], cov[i], edge_y[i], edge_x[i],
                       idx_tf_tg[i], idx_tf_high[i], edge_tg[i], is_high[i], lds);
    }

    // Deterministic block reduction: WMMA wave-sum -> LDS -> WMMA wave-sum.
    const int lane = tid & 31, wid = tid >> 5;
    float w = wave_reduce_wmma(acc);
    if (lane == 0) s_all[SOFF_RED + wid] = w;
    __syncthreads();
    if (tid < 32) {
        float x = (tid < (TPB / 32)) ? s_all[SOFF_RED + tid] : 0.0f;
        float r = wave_reduce_wmma(x);
        if (tid == 0) partials[blockIdx.x] = r;
    }
}

__global__ __launch_bounds__(TPB) void finalize_kernel(
    const float* __restrict__ TFmu,  const float* __restrict__ TFsig,
    const float* __restrict__ TFexp,
    const float* __restrict__ TGmu,  const float* __restrict__ TGsig,
    const float* __restrict__ TGexp, const int* __restrict__ father_num,
    const float* __restrict__ partials, int nblocks,
    int nTFH, int nTG, float* __restrict__ out) {
    __shared__ float red[32];
    const int tid = threadIdx.x;
    float acc = 0.0f;

    // p: sum of top-TF log-probs
    for (int i = tid; i < nTFH; i += TPB) {
        float sig = TFsig[i];
        float z = (TFexp[i] - TFmu[i]) * __builtin_amdgcn_rcpf(sig);
        acc += -0.5f * z * z - __logf(sig) - 0.5f * LOG2PI_F;
    }
    // q: -(father_num-1) weighted TG log-probs
    for (int i = tid; i < nTG; i += TPB) {
        float sig = TGsig[i];
        float z = (TGexp[i] - TGmu[i]) * __builtin_amdgcn_rcpf(sig);
        float lp = -0.5f * z * z - __logf(sig) - 0.5f * LOG2PI_F;
        acc -= ((float)father_num[i] - 1.0f) * lp;
    }
    // edge-kernel partials
    for (int i = tid; i < nblocks; i += TPB) acc += partials[i];

    const int lane = tid & 31, wid = tid >> 5;
    float w = wave_reduce_wmma(acc);
    if (lane == 0) red[wid] = w;
    __syncthreads();
    if (tid < 32) {
        float x = (tid < (TPB / 32)) ? red[tid] : 0.0f;
        float r = wave_reduce_wmma(x);
        if (tid == 0) out[0] = -r;
    }
}

extern "C" void kernel_launch(void* const* d_in, const int* in_sizes, int n_in,
                              void* d_out, int out_size, void* d_ws, size_t ws_size,
                              hipStream_t stream) {
    const float* TF_high_mu    = (const float*)d_in[0];
    const float* TF_high_sigma = (const float*)d_in[1];
    const float* TG_mu         = (const float*)d_in[2];
    const float* TG_sigma      = (const float*)d_in[3];
    const float* TF_high_exp   = (const float*)d_in[4];
    const float* TG_exp        = (const float*)d_in[5];
    const float* k_edge        = (const float*)d_in[6];
    const float* alpha         = (const float*)d_in[7];
    const float* cov           = (const float*)d_in[8];
    const float* edge_y        = (const float*)d_in[9];
    const float* edge_x        = (const float*)d_in[10];
    const int*   father_num    = (const int*)d_in[11];
    const int*   idx_tf_tg     = (const int*)d_in[12];
    const int*   idx_tf_high   = (const int*)d_in[13];
    const int*   edge_tg_idx   = (const int*)d_in[14];
    const int*   is_high       = (const int*)d_in[15];

    const int nTFH = in_sizes[0];
    const int nTG  = in_sizes[2];
    const int nE   = in_sizes[6];

    float* partials = (float*)d_ws;          // NBLOCKS floats

    edge_kernel<<<NBLOCKS, TPB, 0, stream>>>(
        TG_mu, TG_sigma, TF_high_mu, TF_high_sigma,
        k_edge, alpha, cov, edge_y, edge_x,
        idx_tf_tg, idx_tf_high, edge_tg_idx, is_high,
        partials, nE, nTG, nTFH);

    finalize_kernel<<<1, TPB, 0, stream>>>(
        TF_high_mu, TF_high_sigma, TF_high_exp,
        TG_mu, TG_sigma, TG_exp, father_num,
        partials, NBLOCKS, nTFH, nTG, (float*)d_out);
}